// Decoder_49821620633830
// MI455X (gfx1250) — compile-verified
//
#include <hip/hip_runtime.h>

typedef __attribute__((ext_vector_type(2))) float v2f;
typedef __attribute__((ext_vector_type(4))) float v4f;
typedef __attribute__((ext_vector_type(8))) float v8f;

#define IN_DIM   1024
#define HIDDEN   128
#define OUT_DIM  4
#define NN       262144
#define KC       32                 // K-chunk staged in LDS
#define NCHUNK   (IN_DIM / KC)      // 32
#define W1S      144                // LDS row stride in float2 (128 + 16 stagger)
#define ROWS_PER_WG 256             // 8 waves * 32 rows

__global__ __launch_bounds__(256) void mlp_fused_kernel(
    const float* __restrict__ V,
    const float* __restrict__ W1,
    const float* __restrict__ b1,
    const float* __restrict__ W2,
    const float* __restrict__ b2,
    float* __restrict__ out)
{
    // double-buffered W1 K-chunk, packed as (W1[k][n], W1[k+1][n]) pairs
    // row = k-pair index (16 per chunk), col = n (128), stride 144 -> 36,864 B
    __shared__ v2f ldsW1[2][16 * W1S];

    const int tid  = threadIdx.x;
    const int wave = tid >> 5;
    const int lane = tid & 31;
    const int c    = lane & 15;   // column-in-tile / row-in-tile selector
    const int half = lane >> 4;   // K sub-block selector per WMMA layout

    const int rowBase = blockIdx.x * ROWS_PER_WG + wave * 32;

    // ---- W1 staging thread mapping: 256 threads cover 16 kpairs x 128 n
    const int sn = tid & 127;     // n
    const int sp = tid >> 7;      // kpair phase 0..1

    float ra[8], rb[8];
    #pragma unroll
    for (int i = 0; i < 8; ++i) {            // preload chunk 0 into regs
        const int kp = sp + 2 * i;
        const float* g = W1 + (size_t)(2 * kp) * HIDDEN + sn;
        ra[i] = g[0];
        rb[i] = g[HIDDEN];
    }

    v8f acc0[8] = {};   // M-tile 0: rows rowBase..+15, all 128 hidden cols
    v8f acc1[8] = {};   // M-tile 1: rows rowBase+16..+31

    int cur = 0;
    #pragma unroll 1
    for (int chunk = 0; chunk < NCHUNK; ++chunk) {
        // commit staged regs for this chunk to LDS
        #pragma unroll
        for (int i = 0; i < 8; ++i) {
            const int kp = sp + 2 * i;
            v2f p; p.x = ra[i]; p.y = rb[i];
            ldsW1[cur][kp * W1S + sn] = p;
        }
        __syncthreads();

        // issue next chunk's global loads; latency hidden under WMMAs below
        if (chunk + 1 < NCHUNK) {
            const int kc = (chunk + 1) * KC;
            #pragma unroll
            for (int i = 0; i < 8; ++i) {
                const int kp = sp + 2 * i;
                const float* g = W1 + (size_t)(kc + 2 * kp) * HIDDEN + sn;
                ra[i] = g[0];
                rb[i] = g[HIDDEN];
            }
        }

        // ---- consume: A frags stream from global, B frags from LDS
        const float* A0 = V + (size_t)(rowBase + c) * IN_DIM + chunk * KC + 2 * half;
        const float* A1 = A0 + (size_t)16 * IN_DIM;
        const v2f* Bbase = &ldsW1[cur][half * W1S + c];

        #pragma unroll
        for (int kk = 0; kk < KC / 4; ++kk) {
            v2f a0 = *(const v2f*)(A0 + 4 * kk);
            v2f a1 = *(const v2f*)(A1 + 4 * kk);
            const v2f* Brow = Bbase + (2 * kk) * W1S;
            #pragma unroll
            for (int t = 0; t < 8; ++t) {
                v2f b = Brow[16 * t];
                acc0[t] = __builtin_amdgcn_wmma_f32_16x16x4_f32(
                    false, a0, false, b, (short)0, acc0[t], false, false);
                acc1[t] = __builtin_amdgcn_wmma_f32_16x16x4_f32(
                    false, a1, false, b, (short)0, acc1[t], false, false);
            }
        }
        cur ^= 1;
    }

    // ---- epilogue: +b1, ReLU, x W2 [128x4] + b2, in-register
    float b1v[8];
    v4f   w2r[8];
    #pragma unroll
    for (int t = 0; t < 8; ++t) {
        b1v[t] = b1[16 * t + c];
        w2r[t] = ((const v4f*)W2)[16 * t + c];   // W2 row (16t+c), 4 floats
    }
    const float b2v = b2[c & 3];

    #pragma unroll
    for (int mt = 0; mt < 2; ++mt) {
        #pragma unroll
        for (int r = 0; r < 8; ++r) {
            float p0 = 0.f, p1 = 0.f, p2 = 0.f, p3 = 0.f;
            #pragma unroll
            for (int t = 0; t < 8; ++t) {
                const float av = (mt == 0) ? acc0[t][r] : acc1[t][r];
                const float hv = fmaxf(av + b1v[t], 0.0f);
                p0 += hv * w2r[t].x;
                p1 += hv * w2r[t].y;
                p2 += hv * w2r[t].z;
                p3 += hv * w2r[t].w;
            }
            // butterfly-reduce over the 16 N-lanes within each half-wave
            #pragma unroll
            for (int m = 1; m <= 8; m <<= 1) {
                p0 += __shfl_xor(p0, m, 32);
                p1 += __shfl_xor(p1, m, 32);
                p2 += __shfl_xor(p2, m, 32);
                p3 += __shfl_xor(p3, m, 32);
            }
            if (c < OUT_DIM) {
                const float val = (c == 0) ? p0 : (c == 1) ? p1 : (c == 2) ? p2 : p3;
                const int rowg = rowBase + mt * 16 + r + 8 * half;
                out[(size_t)rowg * OUT_DIM + c] = val + b2v;
            }
        }
    }
}

extern "C" void kernel_launch(void* const* d_in, const int* in_sizes, int n_in,
                              void* d_out, int out_size, void* d_ws, size_t ws_size,
                              hipStream_t stream) {
    const float* V  = (const float*)d_in[0];
    const float* W1 = (const float*)d_in[1];
    const float* b1 = (const float*)d_in[2];
    const float* W2 = (const float*)d_in[3];
    const float* b2 = (const float*)d_in[4];
    float* out = (float*)d_out;

    dim3 grid(NN / ROWS_PER_WG);   // 1024 workgroups
    dim3 block(256);               // 8 wave32s
    mlp_fused_kernel<<<grid, block, 0, stream>>>(V, W1, b1, W2, b2, out);
}